// GGNN_52725018526191
// MI455X (gfx1250) — compile-verified
//
#include <hip/hip_runtime.h>
#include <hip/hip_bf16.h>
#include <math.h>

// ---------------- problem constants ----------------
#define Hdim   150
#define H3     450          // 3*H
#define H3P    452          // padded LDS stride for gates
#define Npts   50000
#define Epts   400000
#define Gn     64
#define NSETS  2
#define PASSES 5
#define SLOPE  0.01f
#define KSTEPS 38           // ceil(150/4)
#define ROWT   (Npts/16)    // 3125 row tiles (exact)

typedef __attribute__((ext_vector_type(2))) float v2f;
typedef __attribute__((ext_vector_type(4))) float v4f;
typedef __attribute__((ext_vector_type(8))) float v8f;

__device__ __forceinline__ v2f ld2z(const float* __restrict__ p, bool ok) {
    v2f r;
    if (ok) { r = *(const v2f*)p; } else { r.x = 0.f; r.y = 0.f; }
    return r;
}

__device__ __forceinline__ v8f wmma_f32(v2f a, v2f b, v8f c) {
    // D = A(16x4 f32) * B(4x16 f32) + C(16x16 f32)
    return __builtin_amdgcn_wmma_f32_16x16x4_f32(
        false, a, false, b, (short)0, c, false, false);
}

// ---------------- zero fill ----------------
__global__ void zero_kernel(float* __restrict__ p, long n4) {
    long i = (long)blockIdx.x * blockDim.x + threadIdx.x;
    long stride = (long)gridDim.x * blockDim.x;
    v4f z; z.x = z.y = z.z = z.w = 0.f;
    for (; i < n4; i += stride) ((v4f*)p)[i] = z;
}

// ---------------- T = X @ W^T + b  (N x 150, WMMA f32) ----------------
// blockDim = 128 (4 waves). Each wave owns one 16-row tile, loops 10 col tiles.
__global__ void transform_kernel(const float* __restrict__ X,
                                 const float* __restrict__ W,
                                 const float* __restrict__ bias,
                                 float* __restrict__ T) {
    const int lane = threadIdx.x & 31;
    const int wave = threadIdx.x >> 5;
    const int rowtile = blockIdx.x * 4 + wave;
    if (rowtile >= ROWT) return;                  // wave-uniform exit
    const int row0 = rowtile * 16;
    const int m    = lane & 15;
    const int koff = (lane >> 4) << 1;            // 0 or 2
    const int mb   = (lane >> 4) << 3;            // C-row base offset (0 or 8)

    const float* __restrict__ xrow = X + (long)(row0 + m) * Hdim;
    v2f afr[KSTEPS];
#pragma unroll
    for (int s = 0; s < KSTEPS; ++s) {
        int kk = s * 4 + koff;
        afr[s] = ld2z(xrow + kk, kk < Hdim);
    }

    for (int ct = 0; ct < 10; ++ct) {             // 10*16 = 160 >= 150 cols
        int n = ct * 16 + m;
        const float* __restrict__ wrow = W + (long)(n < Hdim ? n : Hdim - 1) * Hdim;
        v8f acc = {};
#pragma unroll
        for (int s = 0; s < KSTEPS; ++s) {
            int kk = s * 4 + koff;
            v2f bfr = ld2z(wrow + kk, kk < Hdim);
            acc = wmma_f32(afr[s], bfr, acc);
        }
        if (n < Hdim) {
            float bv = bias[n];
#pragma unroll
            for (int r = 0; r < 8; ++r)
                T[(long)(row0 + mb + r) * Hdim + n] = acc[r] + bv;
        }
    }
}

// ---------------- scatter: incoming[dst] += T[src] ----------------
// One wave per edge; lanes cover 75 v2f chunks of the 150-float row.
__global__ void scatter_kernel(const float* __restrict__ T,
                               const int* __restrict__ edges_k,
                               float* __restrict__ incoming) {
    const int lane = threadIdx.x & 31;
    const int wave = threadIdx.x >> 5;
    long e = (long)blockIdx.x * 8 + wave;
    if (e >= Epts) return;
    const int dst = edges_k[e * 2 + 0];
    const int src = edges_k[e * 2 + 1];
    const float* __restrict__ trow = T + (long)src * Hdim;
    float* __restrict__ irow = incoming + (long)dst * Hdim;
    for (int p = lane; p < 75; p += 32) {
        v2f v = *(const v2f*)(trow + 2 * p);
        atomicAdd(irow + 2 * p,     v.x);
        atomicAdd(irow + 2 * p + 1, v.y);
    }
}

// ---------------- fused GRU cell (WMMA f32, one wave / 16-row tile) ----------
__global__ void __launch_bounds__(32)
gru_kernel(const float* __restrict__ inc,   const float* __restrict__ hprev,
           const float* __restrict__ w_ih,  const float* __restrict__ w_hh,
           const float* __restrict__ b_ih,  const float* __restrict__ b_hh,
           float* __restrict__ hnext) {
    __shared__ float gi[16 * H3P];
    __shared__ float gh[16 * H3P];
    const int lane = threadIdx.x;
    const int row0 = blockIdx.x * 16;
    const int m    = lane & 15;
    const int koff = (lane >> 4) << 1;
    const int mb   = (lane >> 4) << 3;

    const float* __restrict__ xi = inc   + (long)(row0 + m) * Hdim;
    const float* __restrict__ xh = hprev + (long)(row0 + m) * Hdim;
    v2f ai[KSTEPS], ah[KSTEPS];
#pragma unroll
    for (int s = 0; s < KSTEPS; ++s) {
        int kk = s * 4 + koff;
        bool ok = kk < Hdim;
        ai[s] = ld2z(xi + kk, ok);
        ah[s] = ld2z(xh + kk, ok);
    }

    for (int ct = 0; ct < 29; ++ct) {             // 29*16 = 464 >= 450 cols
        int n = ct * 16 + m;
        int nc = n < H3 ? n : H3 - 1;
        const float* __restrict__ wi = w_ih + (long)nc * Hdim;
        const float* __restrict__ wh = w_hh + (long)nc * Hdim;
        v8f ci = {}, ch = {};
#pragma unroll
        for (int s = 0; s < KSTEPS; ++s) {
            int kk = s * 4 + koff;
            bool ok = kk < Hdim;
            v2f bi = ld2z(wi + kk, ok);
            v2f bh = ld2z(wh + kk, ok);
            ci = wmma_f32(ai[s], bi, ci);
            ch = wmma_f32(ah[s], bh, ch);
        }
        if (n < H3) {
            float bbi = b_ih[n], bbh = b_hh[n];
#pragma unroll
            for (int r = 0; r < 8; ++r) {
                gi[(mb + r) * H3P + n] = ci[r] + bbi;
                gh[(mb + r) * H3P + n] = ch[r] + bbh;
            }
        }
    }
    __syncthreads();

    // elementwise gating
    for (int r = 0; r < 16; ++r) {
        const float* __restrict__ hrow = hprev + (long)(row0 + r) * Hdim;
        float* __restrict__ orow       = hnext + (long)(row0 + r) * Hdim;
        for (int c = lane; c < Hdim; c += 32) {
            float ir = gi[r * H3P + c],        hr = gh[r * H3P + c];
            float iz = gi[r * H3P + 150 + c],  hz = gh[r * H3P + 150 + c];
            float in = gi[r * H3P + 300 + c],  hn = gh[r * H3P + 300 + c];
            float rg = 1.f / (1.f + expf(-(ir + hr)));
            float zg = 1.f / (1.f + expf(-(iz + hz)));
            float ng = tanhf(in + rg * hn);
            orow[c] = (1.f - zg) * ng + zg * hrow[c];
        }
    }
}

// ---------------- segment sum over graph ids (sorted) ----------------
__global__ void segsum_kernel(const float* __restrict__ nodes,
                              const int* __restrict__ gids,
                              float* __restrict__ gseg) {
    __shared__ float acc[Gn * 152];
    for (int i = threadIdx.x; i < Gn * 152; i += blockDim.x) acc[i] = 0.f;
    __syncthreads();
    const int lane = threadIdx.x & 31;
    const int wave = threadIdx.x >> 5;
    const int n0 = blockIdx.x * 512;
    const int n1 = (n0 + 512 < Npts) ? n0 + 512 : Npts;
    for (int i = n0 + wave; i < n1; i += 8) {
        int g = gids[i];
        const float* __restrict__ row = nodes + (long)i * Hdim;
        for (int c = lane; c < Hdim; c += 32)
            atomicAdd(&acc[g * 152 + c], row[c]);
    }
    __syncthreads();
    for (int i = threadIdx.x; i < Gn * 152; i += blockDim.x) {
        int c = i % 152;
        if (c < Hdim) {
            float v = acc[i];
            if (v != 0.f) atomicAdd(&gseg[(i / 152) * Hdim + c], v);
        }
    }
}

// ---------------- final MLP (single block) ----------------
__device__ __forceinline__ float leaky(float x) { return x > 0.f ? x : SLOPE * x; }

__global__ void mlp_kernel(const float* __restrict__ gseg,
                           const float* __restrict__ pt,
                           const float* __restrict__ fc1_w, const float* __restrict__ fc1_b,
                           const float* __restrict__ fc2_w, const float* __restrict__ fc2_b,
                           const float* __restrict__ fcL_w, const float* __restrict__ fcL_b,
                           float* __restrict__ out) {
    __shared__ float xbuf[Gn * 152];   // x (64 x 151), later reused as a2 (64 x 80)
    __shared__ float a1[Gn * 80];
    // stage x = [relu(nan0(log(gseg))), problem_type]
    for (int i = threadIdx.x; i < Gn * 152; i += blockDim.x) {
        int row = i / 152, c = i % 152;
        float v = 0.f;
        if (c < Hdim) {
            float s = gseg[row * Hdim + c];
            float g = logf(s);
            if (isnan(g)) g = 0.f;
            v = fmaxf(g, 0.f);
        } else if (c == Hdim) {
            v = pt[row];
        }
        xbuf[i] = v;
    }
    __syncthreads();
    // fc1: 151 -> 80
    for (int i = threadIdx.x; i < Gn * 80; i += blockDim.x) {
        int row = i / 80, j = i % 80;
        float s = fc1_b[j];
        for (int c = 0; c < 151; ++c) s += xbuf[row * 152 + c] * fc1_w[j * 151 + c];
        a1[i] = leaky(s);
    }
    __syncthreads();
    // fc2: 80 -> 80 (into xbuf)
    for (int i = threadIdx.x; i < Gn * 80; i += blockDim.x) {
        int row = i / 80, j = i % 80;
        float s = fc2_b[j];
        for (int c = 0; c < 80; ++c) s += a1[row * 80 + c] * fc2_w[j * 80 + c];
        xbuf[i] = leaky(s);
    }
    __syncthreads();
    // fcL: 80 -> 10
    for (int i = threadIdx.x; i < Gn * 10; i += blockDim.x) {
        int row = i / 10, q = i % 10;
        float s = fcL_b[q];
        for (int c = 0; c < 80; ++c) s += xbuf[row * 80 + c] * fcL_w[q * 80 + c];
        out[i] = s;
    }
}

// ---------------- host orchestration ----------------
extern "C" void kernel_launch(void* const* d_in, const int* in_sizes, int n_in,
                              void* d_out, int out_size, void* d_ws, size_t ws_size,
                              hipStream_t stream) {
    const float* nodes_in = (const float*)d_in[0];
    const float* pt       = (const float*)d_in[1];
    const float* edge_W   = (const float*)d_in[2];
    const float* edge_b   = (const float*)d_in[3];
    const float* w_ih     = (const float*)d_in[4];
    const float* w_hh     = (const float*)d_in[5];
    const float* b_ih     = (const float*)d_in[6];
    const float* b_hh     = (const float*)d_in[7];
    const float* fc1_w    = (const float*)d_in[8];
    const float* fc1_b    = (const float*)d_in[9];
    const float* fc2_w    = (const float*)d_in[10];
    const float* fc2_b    = (const float*)d_in[11];
    const float* fcL_w    = (const float*)d_in[12];
    const float* fcL_b    = (const float*)d_in[13];
    const int*   edges    = (const int*)d_in[14];
    const int*   gids     = (const int*)d_in[15];

    float* ws = (float*)d_ws;
    const long NH = (long)Npts * Hdim;
    float* nodesA   = ws;
    float* nodesB   = ws + NH;
    float* incoming = ws + 2 * NH;
    float* T        = ws + 3 * NH;
    float* gseg     = ws + 4 * NH;

    hipMemcpyAsync(nodesA, nodes_in, NH * sizeof(float),
                   hipMemcpyDeviceToDevice, stream);

    float* cur = nodesA;
    float* nxt = nodesB;
    const int tgrid = (ROWT + 3) / 4;          // transform: 4 waves / block
    const int sgrid = Epts / 8;                // scatter:   8 edges / block

    for (int j = 0; j < PASSES; ++j) {
        zero_kernel<<<1024, 256, 0, stream>>>(incoming, NH / 4);
        for (int k = 0; k < NSETS; ++k) {
            transform_kernel<<<tgrid, 128, 0, stream>>>(
                cur, edge_W + (long)k * Hdim * Hdim, edge_b + (long)k * Hdim, T);
            scatter_kernel<<<sgrid, 256, 0, stream>>>(
                T, edges + (long)k * Epts * 2, incoming);
        }
        gru_kernel<<<ROWT, 32, 0, stream>>>(incoming, cur, w_ih, w_hh, b_ih, b_hh, nxt);
        float* t = cur; cur = nxt; nxt = t;
    }

    zero_kernel<<<16, 256, 0, stream>>>(gseg, (long)(Gn * Hdim) / 4);
    segsum_kernel<<<(Npts + 511) / 512, 256, 0, stream>>>(cur, gids, gseg);
    mlp_kernel<<<1, 256, 0, stream>>>(gseg, pt, fc1_w, fc1_b, fc2_w, fc2_b,
                                      fcL_w, fcL_b, (float*)d_out);
}